// AgentAttention_15049565405279
// MI455X (gfx1250) — compile-verified
//
#include <hip/hip_runtime.h>
#include <math.h>

// ---------------------------------------------------------------------------
// Problem constants (from reference)
// ---------------------------------------------------------------------------
#define BATCH   32
#define HH      28
#define WW      28
#define NTOK    784           // HH*WW
#define CDIM    576
#define NHEADS  8
#define HDIM    72            // CDIM / NHEADS
#define NAGENT  49
#define MROWS   (BATCH * NTOK)        // 25088
#define KKAN    (CDIM + CDIM * 8)     // 5184 = silu(576) + bases(4608)

typedef __attribute__((ext_vector_type(2))) float v2f;
typedef __attribute__((ext_vector_type(8))) float v8f;

// ---------------------------------------------------------------------------
// KAN B-spline bases: GRID_SIZE=5, ORDER=3 -> 12 uniform knots at
// g[j] = -2.2 + 0.4*j, 8 output bases per scalar input.
// ---------------------------------------------------------------------------
__device__ __forceinline__ void kan_bases(float x, float* out) {
    float bb[11];
#pragma unroll
    for (int j = 0; j < 11; ++j) {
        float g0 = -2.2f + 0.4f * (float)j;
        bb[j] = (x >= g0 && x < g0 + 0.4f) ? 1.0f : 0.0f;
    }
#pragma unroll
    for (int k = 1; k <= 3; ++k) {
        float inv = 1.0f / (0.4f * (float)k);
#pragma unroll
        for (int j = 0; j + k < 11; ++j) {
            float gj    = -2.2f + 0.4f * (float)j;
            float left  = (x - gj) * inv;
            float right = ((gj + 0.4f * (float)(k + 1)) - x) * inv;
            bb[j] = left * bb[j] + right * bb[j + 1];
        }
    }
#pragma unroll
    for (int kk = 0; kk < 8; ++kk) out[kk] = bb[kk];
}

// ---------------------------------------------------------------------------
// WMMA f32 GEMM: C[M,N] = A' [M,K] x B[K,N].
// MODE 0: A' = A (plain row-major load).
// MODE 2: "KAN virtual A": k < 576  -> silu(A[m,k])
//                          k >= 576 -> bspline_basis_{k&7}(A[m, (k-576)>>3])
//
// Block = 128 threads (4 waves) computes a 64x64 C tile.
// Wave w owns rows [16w, 16w+16) x all 64 cols (4 x v8f accumulators).
// Per 64-wide K block: B(64x64) is staged once into LDS (coalesced reads,
// k-pair-interleaved layout so each lane's B fragment {B[k][n],B[k+1][n]} is
// one 8B-aligned ds_load_b64) and shared by all 4 waves -> 256 wmma per 4096
// B elements; each A fragment (ds_load_b64) feeds 4 wmma.
// Requires: M % 64 == 0, N % 64 == 0, K % 64 == 0 (true for all calls).
// ---------------------------------------------------------------------------
template <int MODE>
__global__ __launch_bounds__(128) void gemm_wmma_kernel(
    const float* __restrict__ A, const float* __restrict__ B,
    float* __restrict__ C, int Nblk, int K, int lda, int ldb, int ldc)
{
    // sB layout: index = (krow>>1)*128 + col*2 + (krow&1)
    __shared__ float sB[64 * 64];        // 16 KB: B K-block, shared by block
    __shared__ float sAall[4][16 * 64];  // 16 KB: per-wave A (or virtual-A)
    const int tid  = threadIdx.x;
    const int wave = tid >> 5;
    const int lane = tid & 31;
    const int nb = blockIdx.x % Nblk, mb = blockIdx.x / Nblk;
    const int n0 = nb * 64;
    const int mrow0 = mb * 64 + wave * 16;
    float* sa = sAall[wave];

    v8f acc[4] = {};
    for (int kb = 0; kb < K; kb += 64) {
        __syncthreads();
        // ---- stage B (64 k x 64 n), coalesced global reads, pair-interleave
#pragma unroll
        for (int t = 0; t < 32; ++t) {
            int idx = tid + 128 * t;
            int row = idx >> 6, col = idx & 63;
            sB[((row >> 1) << 7) + (col << 1) + (row & 1)] =
                B[(size_t)(kb + row) * ldb + n0 + col];
        }
        // ---- stage this wave's A block (16 rows x 64 k) -------------------
        if (MODE == 0 || kb < CDIM) {
#pragma unroll
            for (int t = 0; t < 32; ++t) {
                int idx = lane + 32 * t;
                int row = idx >> 6, kc = idx & 63;
                float v = A[(size_t)(mrow0 + row) * lda + kb + kc];
                if (MODE == 2) v = v / (1.0f + __expf(-v));   // silu region
                sa[idx] = v;
            }
        } else {
            // spline region: 8 input columns per 64-wide K block
#pragma unroll
            for (int t = 0; t < 4; ++t) {
                int pair = lane + 32 * t;            // 0..127 -> (row, i_local)
                int row = pair >> 3, il = pair & 7;
                int icol = ((kb - CDIM) >> 3) + il;
                float xv = A[(size_t)(mrow0 + row) * lda + icol];
                float bs[8];
                kan_bases(xv, bs);
#pragma unroll
                for (int kk = 0; kk < 8; ++kk)
                    sa[row * 64 + il * 8 + kk] = bs[kk];
            }
        }
        __syncthreads();
        if (kb + 64 < K)  // weights stay hot in 192MB L2; hint next block
            __builtin_prefetch(B + (size_t)(kb + 64) * ldb + n0, 0, 1);

        // ---- 16 K-steps x 4 N-subtiles = 64 wmma over this K block --------
#pragma unroll
        for (int ks = 0; ks < 64; ks += 4) {
            const int krow = ks + ((lane >> 4) << 1);   // even
            v2f a = *(const v2f*)&sa[(lane & 15) * 64 + krow];
            const float* bbase = &sB[(krow >> 1) << 7];
#pragma unroll
            for (int ns = 0; ns < 4; ++ns) {
                v2f bf = *(const v2f*)&bbase[(ns * 16 + (lane & 15)) << 1];
                acc[ns] = __builtin_amdgcn_wmma_f32_16x16x4_f32(
                    false, a, false, bf, (short)0, acc[ns], false, false);
            }
        }
    }
#pragma unroll
    for (int ns = 0; ns < 4; ++ns)
#pragma unroll
        for (int r = 0; r < 8; ++r) {
            int row = mrow0 + r + ((lane >> 4) << 3);
            C[(size_t)row * ldc + n0 + ns * 16 + (lane & 15)] = acc[ns][r];
        }
}

// ---------------------------------------------------------------------------
// Weight prep: WqT[k][o]=Wq[o][k]; WkvT[k][j]=Wkv[j][k];
// Bcat (5184 x 576): rows 0..575 = base_weight^T, rows 576.. = (spline*scaler)^T
// ---------------------------------------------------------------------------
__global__ void prep_weights_kernel(
    const float* __restrict__ Wq, const float* __restrict__ Wkv,
    const float* __restrict__ base_w, const float* __restrict__ spl_w,
    const float* __restrict__ spl_s,
    float* __restrict__ WqT, float* __restrict__ WkvT, float* __restrict__ Bcat)
{
    size_t idx = (size_t)blockIdx.x * blockDim.x + threadIdx.x;
    const size_t NQ = (size_t)CDIM * CDIM;
    const size_t NKV = (size_t)CDIM * (2 * CDIM);
    const size_t NB = (size_t)KKAN * CDIM;
    if (idx < NQ) {
        int kk = (int)(idx / CDIM), o = (int)(idx % CDIM);
        WqT[idx] = Wq[(size_t)o * CDIM + kk];
    } else if (idx < NQ + NKV) {
        size_t r = idx - NQ;
        int kk = (int)(r / (2 * CDIM)), j = (int)(r % (2 * CDIM));
        WkvT[r] = Wkv[(size_t)j * CDIM + kk];
    } else if (idx < NQ + NKV + NB) {
        size_t r = idx - NQ - NKV;
        int kidx = (int)(r / CDIM), o = (int)(r % CDIM);
        float v;
        if (kidx < CDIM) {
            v = base_w[(size_t)o * CDIM + kidx];
        } else {
            int t = kidx - CDIM;
            int i = t >> 3, kk = t & 7;
            v = spl_w[((size_t)o * CDIM + i) * 8 + kk] * spl_s[(size_t)o * CDIM + i];
        }
        Bcat[r] = v;
    }
}

// ---------------------------------------------------------------------------
// Positional biases: bilinear 7x7 -> 28x28 (half-pixel centers, edge clamp)
// pb1/ab1 stored [hh, a, n]; pb2 [hh, a, n]; ab2 [hh, n, a]
// ---------------------------------------------------------------------------
__device__ __forceinline__ float bilin7(const float* img, float fy, float fx) {
    int y0 = (int)floorf(fy), x0 = (int)floorf(fx);
    float wy = fy - (float)y0, wx = fx - (float)x0;
    int y1 = y0 + 1, x1 = x0 + 1;
    y0 = min(max(y0, 0), 6); y1 = min(max(y1, 0), 6);
    x0 = min(max(x0, 0), 6); x1 = min(max(x1, 0), 6);
    float v00 = img[y0 * 7 + x0], v01 = img[y0 * 7 + x1];
    float v10 = img[y1 * 7 + x0], v11 = img[y1 * 7 + x1];
    return v00 * (1.f - wy) * (1.f - wx) + v01 * (1.f - wy) * wx +
           v10 * wy * (1.f - wx) + v11 * wy * wx;
}

__global__ void bias_kernel(
    const float* __restrict__ an_bias, const float* __restrict__ na_bias,
    const float* __restrict__ ah_bias, const float* __restrict__ aw_bias,
    const float* __restrict__ ha_bias, const float* __restrict__ wa_bias,
    float* __restrict__ pb1, float* __restrict__ pb2,
    float* __restrict__ ab1, float* __restrict__ ab2)
{
    int idx = blockIdx.x * blockDim.x + threadIdx.x;  // NHEADS*NAGENT*NTOK
    if (idx >= NHEADS * NAGENT * NTOK) return;
    int x = idx % WW, y = (idx / WW) % HH;
    int a = (idx / NTOK) % NAGENT, hh = idx / (NTOK * NAGENT);
    float fy = ((float)y + 0.5f) * 0.25f - 0.5f;
    float fx = ((float)x + 0.5f) * 0.25f - 0.5f;
    const float* i1 = an_bias + ((size_t)hh * NAGENT + a) * 49;
    const float* i2 = na_bias + ((size_t)hh * NAGENT + a) * 49;
    pb1[idx] = bilin7(i1, fy, fx);
    ab1[idx] = bilin7(i2, fy, fx);
    pb2[idx] = ah_bias[((size_t)hh * NAGENT + a) * HH + y] +
               aw_bias[((size_t)hh * NAGENT + a) * WW + x];
    ab2[((size_t)hh * NTOK + y * WW + x) * NAGENT + a] =
        ha_bias[((size_t)hh * HH + y) * NAGENT + a] +
        wa_bias[((size_t)hh * WW + x) * NAGENT + a];
}

// ---------------------------------------------------------------------------
// Agent tokens: 4x4 mean pool of q on the 28x28 grid -> (b, 49, c)
// ---------------------------------------------------------------------------
__global__ void agent_pool_kernel(const float* __restrict__ q,
                                  float* __restrict__ agent)
{
    int idx = blockIdx.x * blockDim.x + threadIdx.x;  // BATCH*NAGENT*CDIM
    if (idx >= BATCH * NAGENT * CDIM) return;
    int cd = idx % CDIM;
    int t = idx / CDIM;
    int a = t % NAGENT, batch = t / NAGENT;
    int ap = a / 7, aq = a % 7;
    float s = 0.f;
#pragma unroll
    for (int i = 0; i < 4; ++i)
#pragma unroll
        for (int j = 0; j < 4; ++j) {
            int y = ap * 4 + i, x = aq * 4 + j;
            s += q[((size_t)batch * NTOK + y * WW + x) * CDIM + cd];
        }
    agent[idx] = s * (1.0f / 16.0f);
}

// ---------------------------------------------------------------------------
// agent_attn: per (batch, head, agent): softmax_n( (ah*scale)·k + pb1 + pb2 ),
// then agent_v = attn @ v. One 256-thread block per (b,h,a).
// ---------------------------------------------------------------------------
__global__ __launch_bounds__(256) void agent_attn_kernel(
    const float* __restrict__ agent, const float* __restrict__ kvb,
    const float* __restrict__ pb1, const float* __restrict__ pb2,
    float* __restrict__ agent_v)
{
    int a = blockIdx.x % NAGENT;
    int hh = (blockIdx.x / NAGENT) % NHEADS;
    int batch = blockIdx.x / (NAGENT * NHEADS);
    __shared__ float sA[HDIM];
    __shared__ float probs[NTOK];
    __shared__ float rbuf[256];
    const int tid = threadIdx.x;
    const float scale = 0.1178511301977579f;   // 72^-0.5

    for (int d = tid; d < HDIM; d += 256)
        sA[d] = agent[((size_t)batch * NAGENT + a) * CDIM + hh * HDIM + d] * scale;
    __syncthreads();

    float lmax = -3.4e38f;
    for (int ni = tid; ni < NTOK; ni += 256) {
        const float* kp = kvb + ((size_t)batch * NTOK + ni) * (2 * CDIM) + hh * HDIM;
        float s = 0.f;
#pragma unroll 8
        for (int d = 0; d < HDIM; ++d) s += sA[d] * kp[d];
        size_t bo = ((size_t)hh * NAGENT + a) * NTOK + ni;
        s += pb1[bo] + pb2[bo];
        probs[ni] = s;
        lmax = fmaxf(lmax, s);
    }
    rbuf[tid] = lmax; __syncthreads();
    for (int s = 128; s > 0; s >>= 1) {
        if (tid < s) rbuf[tid] = fmaxf(rbuf[tid], rbuf[tid + s]);
        __syncthreads();
    }
    float gmax = rbuf[0];
    __syncthreads();

    float lsum = 0.f;
    for (int ni = tid; ni < NTOK; ni += 256) {
        float e = __expf(probs[ni] - gmax);
        probs[ni] = e;
        lsum += e;
    }
    rbuf[tid] = lsum; __syncthreads();
    for (int s = 128; s > 0; s >>= 1) {
        if (tid < s) rbuf[tid] += rbuf[tid + s];
        __syncthreads();
    }
    float inv = 1.f / rbuf[0];
    __syncthreads();

    for (int d = tid; d < HDIM; d += 256) {
        float accv = 0.f;
        for (int ni = 0; ni < NTOK; ++ni)
            accv += probs[ni] *
                    kvb[((size_t)batch * NTOK + ni) * (2 * CDIM) + CDIM + hh * HDIM + d];
        agent_v[(((size_t)batch * NHEADS + hh) * NAGENT + a) * HDIM + d] = accv * inv;
    }
}

// ---------------------------------------------------------------------------
// q_attn: per token softmax over 49 agents, then mix agent_v.
// Block = 128 threads = one (b, h, 112-token chunk); agent tiles in LDS.
// ---------------------------------------------------------------------------
__global__ __launch_bounds__(128) void q_attn_kernel(
    const float* __restrict__ qbuf, const float* __restrict__ agent,
    const float* __restrict__ agent_v, const float* __restrict__ ab1,
    const float* __restrict__ ab2, float* __restrict__ outb)
{
    int chunk = blockIdx.x % 7;
    int hh = (blockIdx.x / 7) % NHEADS;
    int batch = blockIdx.x / (7 * NHEADS);
    __shared__ float agA[NAGENT * HDIM];
    __shared__ float agV[NAGENT * HDIM];
    const int tid = threadIdx.x;
    const float scale = 0.1178511301977579f;

    for (int idx = tid; idx < NAGENT * HDIM; idx += 128) {
        int a = idx / HDIM, d = idx % HDIM;
        agA[idx] = agent[((size_t)batch * NAGENT + a) * CDIM + hh * HDIM + d] * scale;
        agV[idx] = agent_v[(((size_t)batch * NHEADS + hh) * NAGENT + a) * HDIM + d];
    }
    __syncthreads();
    if (tid >= 112) return;
    const int ni = chunk * 112 + tid;

    const float* qp = qbuf + ((size_t)batch * NTOK + ni) * CDIM + hh * HDIM;
    float qr[HDIM];
#pragma unroll
    for (int d = 0; d < HDIM; ++d) qr[d] = qp[d];

    float s[NAGENT];
    float smax = -3.4e38f;
    for (int a = 0; a < NAGENT; ++a) {
        float acc = 0.f;
#pragma unroll 8
        for (int d = 0; d < HDIM; ++d) acc += qr[d] * agA[a * HDIM + d];
        acc += ab1[((size_t)hh * NAGENT + a) * NTOK + ni] +
               ab2[((size_t)hh * NTOK + ni) * NAGENT + a];
        s[a] = acc;
        smax = fmaxf(smax, acc);
    }
    float ssum = 0.f;
    for (int a = 0; a < NAGENT; ++a) {
        float e = __expf(s[a] - smax);
        s[a] = e;
        ssum += e;
    }
    float inv = 1.f / ssum;
    float* op = outb + ((size_t)batch * NTOK + ni) * CDIM + hh * HDIM;
    for (int d = 0; d < HDIM; ++d) {
        float acc = 0.f;
        for (int a = 0; a < NAGENT; ++a) acc += s[a] * agV[a * HDIM + d];
        op[d] = acc * inv;
    }
}

// ---------------------------------------------------------------------------
// Depthwise 3x3 conv on v image + BN + ReLU, added into attn output.
// ---------------------------------------------------------------------------
__global__ void dwconv_kernel(
    const float* __restrict__ kvb, const float* __restrict__ dwc_w,
    const float* __restrict__ dwc_b, const float* __restrict__ gamma,
    const float* __restrict__ beta, const float* __restrict__ mean,
    const float* __restrict__ var, float* __restrict__ attn)
{
    size_t idx = (size_t)blockIdx.x * blockDim.x + threadIdx.x;
    if (idx >= (size_t)MROWS * CDIM) return;
    int cd = (int)(idx % CDIM);
    int pix = (int)((idx / CDIM) % NTOK);
    int batch = (int)(idx / ((size_t)CDIM * NTOK));
    int y = pix / WW, x = pix % WW;
    float acc = dwc_b[cd];
#pragma unroll
    for (int ky = 0; ky < 3; ++ky)
#pragma unroll
        for (int kx = 0; kx < 3; ++kx) {
            int sy = y + ky - 1, sx = x + kx - 1;
            if (sy >= 0 && sy < HH && sx >= 0 && sx < WW)
                acc += kvb[((size_t)batch * NTOK + sy * WW + sx) * (2 * CDIM) + CDIM + cd] *
                       dwc_w[(size_t)cd * 9 + ky * 3 + kx];
        }
    float sc = gamma[cd] * rsqrtf(var[cd] + 1e-5f);
    float yv = (acc - mean[cd]) * sc + beta[cd];
    attn[idx] += fmaxf(yv, 0.f);
}

// ---------------------------------------------------------------------------
// Launcher
// ---------------------------------------------------------------------------
extern "C" void kernel_launch(void* const* d_in, const int* in_sizes, int n_in,
                              void* d_out, int out_size, void* d_ws, size_t ws_size,
                              hipStream_t stream) {
    const float* x       = (const float*)d_in[0];
    const float* Wq      = (const float*)d_in[3];
    const float* Wkv     = (const float*)d_in[4];
    const float* an_bias = (const float*)d_in[5];
    const float* na_bias = (const float*)d_in[6];
    const float* ah_bias = (const float*)d_in[7];
    const float* aw_bias = (const float*)d_in[8];
    const float* ha_bias = (const float*)d_in[9];
    const float* wa_bias = (const float*)d_in[10];
    const float* dwc_w   = (const float*)d_in[11];
    const float* dwc_b   = (const float*)d_in[12];
    const float* bn_g    = (const float*)d_in[13];
    const float* bn_b    = (const float*)d_in[14];
    const float* bn_m    = (const float*)d_in[15];
    const float* bn_v    = (const float*)d_in[16];
    const float* base_w  = (const float*)d_in[17];
    const float* spl_w   = (const float*)d_in[18];
    const float* spl_s   = (const float*)d_in[19];
    float* out = (float*)d_out;
    float* ws  = (float*)d_ws;

    size_t off = 0;
    float* qbuf = ws + off;  off += (size_t)MROWS * CDIM;           // 14.45M
    float* kvb  = ws + off;  off += (size_t)MROWS * 2 * CDIM;       // 28.90M
    float* agnt = ws + off;  off += (size_t)BATCH * NAGENT * CDIM;  // 0.90M
    float* pb1  = ws + off;  off += (size_t)NHEADS * NAGENT * NTOK;
    float* pb2  = ws + off;  off += (size_t)NHEADS * NAGENT * NTOK;
    float* ab1  = ws + off;  off += (size_t)NHEADS * NAGENT * NTOK;
    float* ab2  = ws + off;  off += (size_t)NHEADS * NAGENT * NTOK;
    float* agv  = ws + off;  off += (size_t)BATCH * NHEADS * NAGENT * HDIM;
    float* attn = ws + off;  off += (size_t)MROWS * CDIM;           // 14.45M
    float* WqT  = ws + off;  off += (size_t)CDIM * CDIM;
    float* WkvT = ws + off;  off += (size_t)CDIM * 2 * CDIM;
    float* Bcat = ws + off;  off += (size_t)KKAN * CDIM;

    // 1) weight prep + bias precompute
    {
        size_t total = (size_t)CDIM * CDIM + (size_t)CDIM * 2 * CDIM + (size_t)KKAN * CDIM;
        prep_weights_kernel<<<(unsigned)((total + 255) / 256), 256, 0, stream>>>(
            Wq, Wkv, base_w, spl_w, spl_s, WqT, WkvT, Bcat);
    }
    bias_kernel<<<(NHEADS * NAGENT * NTOK + 255) / 256, 256, 0, stream>>>(
        an_bias, na_bias, ah_bias, aw_bias, ha_bias, wa_bias, pb1, pb2, ab1, ab2);

    // 2) projections: q = x@WqT, kv = x@WkvT  (WMMA f32, 64x64 C tiles)
    const int MB = MROWS / 64;                       // 392
    gemm_wmma_kernel<0><<<MB * (CDIM / 64), 128, 0, stream>>>(
        x, WqT, qbuf, CDIM / 64, CDIM, CDIM, CDIM, CDIM);
    gemm_wmma_kernel<0><<<MB * (2 * CDIM / 64), 128, 0, stream>>>(
        x, WkvT, kvb, 2 * CDIM / 64, CDIM, CDIM, 2 * CDIM, 2 * CDIM);

    // 3) attention
    agent_pool_kernel<<<(BATCH * NAGENT * CDIM + 255) / 256, 256, 0, stream>>>(qbuf, agnt);
    agent_attn_kernel<<<BATCH * NHEADS * NAGENT, 256, 0, stream>>>(agnt, kvb, pb1, pb2, agv);
    q_attn_kernel<<<BATCH * NHEADS * 7, 128, 0, stream>>>(qbuf, agnt, agv, ab1, ab2, attn);
    {
        size_t total = (size_t)MROWS * CDIM;
        dwconv_kernel<<<(unsigned)((total + 255) / 256), 256, 0, stream>>>(
            kvb, dwc_w, dwc_b, bn_g, bn_b, bn_m, bn_v, attn);
    }

    // 4) fused KAN: out = [silu(attn), bsplines(attn)] @ Bcat  (K = 5184, WMMA f32)
    gemm_wmma_kernel<2><<<MB * (CDIM / 64), 128, 0, stream>>>(
        attn, Bcat, out, CDIM / 64, KKAN, CDIM, CDIM, CDIM);
}